// GraphConvDistanceLayer_70016556859564
// MI455X (gfx1250) — compile-verified
//
#include <hip/hip_runtime.h>
#include <hip/hip_bf16.h>
#include <stdint.h>

// Problem constants (match reference).
#define N_OUT   262144
#define E_NNZ   16777216
#define NPARAMS 1280
#define NB      16

typedef int      v4i   __attribute__((ext_vector_type(4)));
typedef float    v4f   __attribute__((ext_vector_type(4)));
typedef uint32_t u32x4 __attribute__((ext_vector_type(4)));
typedef int32_t  i32x8 __attribute__((ext_vector_type(8)));
typedef int32_t  i32x4 __attribute__((ext_vector_type(4)));

// ---------------------------------------------------------------------------
// Kernel 1: out[i] = Param_b[b_params[i]]   (4-wide; also clears poison)
// ---------------------------------------------------------------------------
__global__ __launch_bounds__(256) void bias_init_kernel(
    const float* __restrict__ Pb, const v4i* __restrict__ bidx4,
    v4f* __restrict__ out4, int nvec) {
  int i = blockIdx.x * blockDim.x + threadIdx.x;
  if (i < nvec) {
    v4i b = __builtin_nontemporal_load(&bidx4[i]);   // stream-once indices
    v4f o;
    o.x = Pb[b.x];                                   // Pb: 64 B, cache-resident
    o.y = Pb[b.y];
    o.z = Pb[b.z];
    o.w = Pb[b.w];
    out4[i] = o;                                     // b128 store
  }
}

// ---------------------------------------------------------------------------
// Kernel 2: for each (r,c,p): out[r] += sW[p] * x[c]
//   - index streams read as NT int4 (stream-once, keep L2 for x/out)
//   - Param_W staged to LDS via the Tensor Data Mover (gfx1250 TDM)
//   - native no-return global_atomic_add_f32 scatter
// ---------------------------------------------------------------------------
__global__ __launch_bounds__(256) void scatter_scale_kernel(
    const float* __restrict__ x, const float* __restrict__ Pw,
    const v4i* __restrict__ rows4, const v4i* __restrict__ cols4,
    const v4i* __restrict__ params4, float* __restrict__ out, int nvec) {
  __shared__ float sW[NPARAMS];  // 5120 B

#if __has_builtin(__builtin_amdgcn_tensor_load_to_lds) && \
    __has_builtin(__builtin_amdgcn_s_wait_tensorcnt)
  // One wave per block issues a 1-D TDM load of Param_W into LDS.
  if ((threadIdx.x >> 5) == 0) {
    uint64_t ga  = (uint64_t)(uintptr_t)Pw;            // global addr of tile
    uint32_t lds = (uint32_t)(uintptr_t)&sW[0];        // LDS byte offset (addr[31:0])

    u32x4 g0;
    g0[0] = 1u;                                        // count=1, user descriptor
    g0[1] = lds;                                       // lds_addr [63:32]
    g0[2] = (uint32_t)ga;                              // global_addr[31:0]
    g0[3] = (uint32_t)((ga >> 32) & 0x01FFFFFFu)       // global_addr[56:32]
          | (2u << 30);                                // type = 2 ("image")

    i32x8 g1;
    g1[0] = (int32_t)(2u << 16);                       // data_size=2 (4 bytes), mask=0
    g1[1] = (int32_t)((uint32_t)NPARAMS << 16);        // tensor_dim0[15:0] in [31:16]
    g1[2] = (int32_t)(1u << 16);                       // tensor_dim0[31:16]=0; tensor_dim1=1
    g1[3] = (int32_t)((uint32_t)NPARAMS << 16);        // tile_dim0 = 1280
    g1[4] = 1;                                         // tile_dim1 = 1, tile_dim2 = 0
    g1[5] = NPARAMS;                                   // tensor_dim0_stride[31:0]
    g1[6] = 0;                                         // stride[47:32]=0, dim1_stride lo=0
    g1[7] = 0;

    i32x4 z4 = {0, 0, 0, 0};                           // groups 2/3 unused (<=2-D)
    i32x8 z8 = {0, 0, 0, 0, 0, 0, 0, 0};               // trailing group (clang-23 form)
    __builtin_amdgcn_tensor_load_to_lds(g0, g1, z4, z4, z8, /*cpol=*/0);
    __builtin_amdgcn_s_wait_tensorcnt(0);              // s_wait_tensorcnt 0
  }
#else
  for (int k = (int)threadIdx.x; k < NPARAMS; k += (int)blockDim.x) sW[k] = Pw[k];
#endif
  __syncthreads();

  const int tid    = (int)(blockIdx.x * blockDim.x + threadIdx.x);
  const int stride = (int)(gridDim.x * blockDim.x);

  for (int i = tid; i < nvec; i += stride) {
    int nx = i + stride;                               // prefetch next strided chunk
    if (nx < nvec) {
      __builtin_prefetch(&rows4[nx],   0, 0);          // global_prefetch_b8
      __builtin_prefetch(&cols4[nx],   0, 0);
      __builtin_prefetch(&params4[nx], 0, 0);
    }

    v4i r = __builtin_nontemporal_load(&rows4[i]);     // TH=NT b128 streams
    v4i c = __builtin_nontemporal_load(&cols4[i]);
    v4i p = __builtin_nontemporal_load(&params4[i]);

    float v0 = sW[p.x] * x[c.x];
    float v1 = sW[p.y] * x[c.y];
    float v2 = sW[p.z] * x[c.z];
    float v3 = sW[p.w] * x[c.w];

    unsafeAtomicAdd(&out[r.x], v0);                    // global_atomic_add_f32 (no-rtn)
    unsafeAtomicAdd(&out[r.y], v1);
    unsafeAtomicAdd(&out[r.z], v2);
    unsafeAtomicAdd(&out[r.w], v3);
  }
}

// ---------------------------------------------------------------------------
// Host-side launcher
// Inputs (setup_inputs order):
//   0: x        [N]   f32      4: w_cols   [E] i32
//   1: Param_W  [1280] f32     5: w_params [E] i32
//   2: Param_b  [16]  f32      6: b_params [N] i32
//   3: w_rows   [E]   i32
// Output: [N] f32
// ---------------------------------------------------------------------------
extern "C" void kernel_launch(void* const* d_in, const int* in_sizes, int n_in,
                              void* d_out, int out_size, void* d_ws, size_t ws_size,
                              hipStream_t stream) {
  const float* x        = (const float*)d_in[0];
  const float* Pw       = (const float*)d_in[1];
  const float* Pb       = (const float*)d_in[2];
  const int*   w_rows   = (const int*)d_in[3];
  const int*   w_cols   = (const int*)d_in[4];
  const int*   w_params = (const int*)d_in[5];
  const int*   b_params = (const int*)d_in[6];
  float*       out      = (float*)d_out;

  const int n     = out_size;            // N (multiple of 4)
  const int e     = in_sizes[3];         // E
  const int nvec  = e >> 2;              // int4 packets over E
  const int nbias = n >> 2;              // float4 packets over N

  // 1) bias init (overwrites poison; stream-serialized before scatter).
  bias_init_kernel<<<(nbias + 255) / 256, 256, 0, stream>>>(
      Pb, (const v4i*)b_params, (v4f*)out, nbias);

  // 2) scatter-add. 2048 blocks x 256 thr (8 waves/block under wave32):
  //    524288 threads, ~8 int4 packets each over nvec = 4194304.
  scatter_scale_kernel<<<2048, 256, 0, stream>>>(
      x, Pw, (const v4i*)w_rows, (const v4i*)w_cols, (const v4i*)w_params,
      out, nvec);
}